// TeamMatchupModel_74217034875090
// MI455X (gfx1250) — compile-verified
//
#include <hip/hip_runtime.h>
#include <math.h>

typedef __attribute__((ext_vector_type(2))) float v2f;
typedef __attribute__((ext_vector_type(8))) float v8f;

#define TEAM_T 6
#define EMB_D  64
#define HID    128
#define XS     132   // x-tile row stride in floats (bank-conflict-free A frags)
#define WS     144   // weight row stride in floats (2*WS % 64 == 32 -> disjoint halves)
#define ROWS_PER_WAVE 16
#define WAVES   8
#define THREADS 256
#define ROWS_PER_BLOCK (ROWS_PER_WAVE * WAVES)

// One 128x128 f32 GEMM (x[16x128] @ W[128x128] + bias), ReLU, written back
// in-place into the wave's x-tile. A fragments are fully hoisted to registers
// first, so the in-place store is hazard-free. Bias-splat accumulator init
// acts as a scheduling fence per tile: it keeps the DS-load lookahead bounded
// so register pressure stays under 256 (no vgpr_msb, no scratch spills),
// while the full nt unroll still exposes 8 independent WMMA chains.
__device__ __forceinline__ void gemm128_relu_inplace(float* __restrict__ xr,
                                                     const float* __restrict__ wbuf,
                                                     const float* __restrict__ bias,
                                                     int sub, int half) {
  const int kb = half * 2;  // lanes 16-31 hold K+2 / K+3 of each 4-wide K chunk
  v2f afrag[32];
#pragma unroll
  for (int k = 0; k < 32; ++k) {
    const float* p = xr + sub * XS + k * 4 + kb;
    afrag[k].x = p[0];
    afrag[k].y = p[1];
  }
  for (int nt = 0; nt < 8; ++nt) {
    const int n = nt * 16 + sub;       // output column for this lane
    const float bn = bias[n];
    v8f c;
#pragma unroll
    for (int i = 0; i < 8; ++i) c[i] = bn;
#pragma unroll
    for (int k = 0; k < 32; ++k) {
      const float* wp = wbuf + (k * 4 + kb) * WS + n;
      v2f bf;
      bf.x = wp[0];
      bf.y = wp[WS];
      c = __builtin_amdgcn_wmma_f32_16x16x4_f32(false, afrag[k], false, bf,
                                                (short)0, c, false, false);
    }
#pragma unroll
    for (int i = 0; i < 8; ++i)
      xr[(half * 8 + i) * XS + n] = fmaxf(c[i], 0.0f);
  }
}

__device__ __forceinline__ void stage_weights(float* __restrict__ wbuf,
                                              const float* __restrict__ W,
                                              int tid) {
  const float4* src = reinterpret_cast<const float4*>(W);
  for (int i = tid; i < HID * (HID / 4); i += THREADS) {
    const int row = i >> 5;   // 32 float4 per 128-float row
    const int c4  = i & 31;
    float4 v = src[i];
    float* d = wbuf + row * WS + c4 * 4;
    d[0] = v.x; d[1] = v.y; d[2] = v.z; d[3] = v.w;
  }
}

__launch_bounds__(THREADS)
__global__ void team_matchup_fused_kernel(
    const int* __restrict__ aIdx, const int* __restrict__ bIdx,
    const int* __restrict__ aLen, const int* __restrict__ bLen,
    const float* __restrict__ emb,
    const float* __restrict__ W1, const float* __restrict__ b1,
    const float* __restrict__ W2, const float* __restrict__ b2,
    const float* __restrict__ W3, const float* __restrict__ b3,
    float* __restrict__ out) {
  extern __shared__ float smem[];
  float* wbuf = smem;                 // HID * WS  (W1, then W2)
  float* xall = smem + HID * WS;      // WAVES * 16 * XS

  const int tid  = threadIdx.x;
  const int lane = tid & 31;
  const int wave = tid >> 5;
  const int sub  = lane & 15;
  const int half = lane >> 4;         // 0: team A / rows 0-7, 1: team B / rows 8-15
  const int tileBase = blockIdx.x * ROWS_PER_BLOCK + wave * ROWS_PER_WAVE;
  float* xr = xall + wave * (ROWS_PER_WAVE * XS);

  // ---- warm L2 with W2 while we're HBM-bound on the gather ----
  for (int i = tid * 32; i < HID * HID; i += THREADS * 32)
    __builtin_prefetch(W2 + i, 0, 1);           // global_prefetch_b8

  // ---- stage W1 into LDS while gathering embeddings ----
  stage_weights(wbuf, W1, tid);

  // ---- gather + masked mean: lanes 0-15 team A, lanes 16-31 team B ----
  {
    const int* idxBase = half ? bIdx : aIdx;
    const int* lenBase = half ? bLen : aLen;
    for (int r = 0; r < ROWS_PER_WAVE; ++r) {
      const int g = tileBase + r;
      const int len = lenBase[g];
      const int* ip = idxBase + g * TEAM_T;
      float ax = 0.f, ay = 0.f, az = 0.f, aw = 0.f;
#pragma unroll
      for (int t = 0; t < TEAM_T; ++t) {
        if (t < len) {
          const long long id = ip[t];
          const float4 e =
              reinterpret_cast<const float4*>(emb + id * (long long)EMB_D)[sub];
          ax += e.x; ay += e.y; az += e.z; aw += e.w;
        }
      }
      const float inv = 1.0f / (float)len;
      float* d = xr + r * XS + half * EMB_D + sub * 4;
      d[0] = ax * inv; d[1] = ay * inv; d[2] = az * inv; d[3] = aw * inv;
    }
  }
  __syncthreads();

  // ---- layer 1: h1 = relu(x @ W1 + b1), in-place in x tile ----
  gemm128_relu_inplace(xr, wbuf, b1, sub, half);
  __syncthreads();                    // everyone done reading W1 + h1 visible

  // ---- restage W2 (already hot in L2 from the prefetch) ----
  stage_weights(wbuf, W2, tid);
  __syncthreads();

  // ---- layer 2: h2 = relu(h1 @ W2 + b2) ----
  gemm128_relu_inplace(xr, wbuf, b2, sub, half);
  __syncthreads();                    // h2 visible across lanes

  // ---- output: sigmoid(h2 @ W3 + b3), two lanes per row ----
  {
    float s = 0.0f;
    const float2* hrow =
        reinterpret_cast<const float2*>(xr + sub * XS + half * EMB_D);
    const float2* w3 = reinterpret_cast<const float2*>(W3 + half * EMB_D);
#pragma unroll
    for (int c = 0; c < EMB_D / 2; ++c) {
      const float2 h2v = hrow[c];
      const float2 wv  = w3[c];
      s += h2v.x * wv.x + h2v.y * wv.y;
    }
    s += __shfl_xor(s, 16, 32);       // fold team-B half into lanes 0-15
    if (half == 0) {
      const float z = s + b3[0];
      out[tileBase + sub] = 1.0f / (1.0f + __expf(-z));
    }
  }
}

extern "C" void kernel_launch(void* const* d_in, const int* in_sizes, int n_in,
                              void* d_out, int out_size, void* d_ws, size_t ws_size,
                              hipStream_t stream) {
  const int*   aIdx = (const int*)d_in[0];
  const int*   bIdx = (const int*)d_in[1];
  const int*   aLen = (const int*)d_in[2];
  const int*   bLen = (const int*)d_in[3];
  const float* emb  = (const float*)d_in[4];
  const float* W1   = (const float*)d_in[5];
  const float* b1   = (const float*)d_in[6];
  const float* W2   = (const float*)d_in[7];
  const float* b2   = (const float*)d_in[8];
  const float* W3   = (const float*)d_in[9];
  const float* b3   = (const float*)d_in[10];
  float* out = (float*)d_out;

  const int B = in_sizes[2];  // a_len has one entry per batch row
  const int grid = (B + ROWS_PER_BLOCK - 1) / ROWS_PER_BLOCK;
  const size_t smem_bytes =
      (size_t)(HID * WS + WAVES * ROWS_PER_WAVE * XS) * sizeof(float);

  // Opt in to >64KB dynamic LDS (gfx1250 WGP supports 320KB).
  (void)hipFuncSetAttribute((const void*)team_matchup_fused_kernel,
                            hipFuncAttributeMaxDynamicSharedMemorySize,
                            (int)smem_bytes);

  team_matchup_fused_kernel<<<grid, THREADS, smem_bytes, stream>>>(
      aIdx, bIdx, aLen, bLen, emb, W1, b1, W2, b2, W3, b3, out);
}